// LightConv1d_3272765079873
// MI455X (gfx1250) — compile-verified
//
#include <hip/hip_runtime.h>
#include <hip/hip_bf16.h>

// Lightweight conv (softmax-normalized depthwise conv1d), B=8 C=1024 T=4096 H=16 K=7,
// causal pad 6. Memory-bound (AI ~1.75 flop/B): optimize pure streaming via
// async global->LDS (CDNA5 GLOBAL_LOAD_ASYNC_TO_LDS_B128) + NT stores.

#define LC_B       8
#define LC_C       1024
#define LC_T       4096
#define LC_H       16
#define LC_K       7
#define LC_HALO    8       // >= K-1, float4-aligned left pad in LDS
#define LC_THREADS 256
#define LC_ITERS   4       // (LC_T/4) / LC_THREADS float4 chunks per thread

typedef float v4f __attribute__((ext_vector_type(4)));
typedef int   v4i __attribute__((ext_vector_type(4)));

#if defined(__gfx1250__) && __has_builtin(__builtin_amdgcn_global_load_async_to_lds_b128)
#define LC_HAVE_ASYNC 1
#else
#define LC_HAVE_ASYNC 0
#endif

static __device__ __forceinline__ void lc_wait_async() {
#if defined(__gfx1250__) && LC_HAVE_ASYNC
#if __has_builtin(__builtin_amdgcn_s_wait_asynccnt)
  __builtin_amdgcn_s_wait_asynccnt(0);
#else
  asm volatile("s_wait_asynccnt 0" ::: "memory");
#endif
#endif
}

// ---- kernel 1: softmax over K for each of H heads -> d_ws ------------------
extern "C" __global__ void lc_softmax_kernel(const float* __restrict__ w,
                                             float* __restrict__ sw) {
  const int h = threadIdx.x;
  if (h < LC_H) {
    float v[LC_K];
    float m = -3.4e38f;
#pragma unroll
    for (int k = 0; k < LC_K; ++k) {
      v[k] = w[h * LC_K + k];
      m = fmaxf(m, v[k]);
    }
    float s = 0.0f;
#pragma unroll
    for (int k = 0; k < LC_K; ++k) {
      v[k] = __expf(v[k] - m);
      s += v[k];
    }
    const float inv = 1.0f / s;
#pragma unroll
    for (int k = 0; k < LC_K; ++k) sw[h * LC_K + k] = v[k] * inv;
  }
}

// ---- kernel 2: one block per (b,c) row; stream via LDS ---------------------
extern "C" __global__ void __launch_bounds__(LC_THREADS)
lc_conv_kernel(const float* __restrict__ x,
               const float* __restrict__ sw,
               const float* __restrict__ bias,
               float* __restrict__ out) {
  __shared__ __align__(16) float tile[LC_HALO + LC_T];

  const int row = blockIdx.x;           // 0 .. B*C-1
  const int c   = row & (LC_C - 1);
  const int h   = c & (LC_H - 1);       // head = c mod 16 (reshape is h-inner)
  const int tid = threadIdx.x;

  const float* __restrict__ xrow = x + (size_t)row * LC_T;
  float* __restrict__ orow       = out + (size_t)row * LC_T;

  // zero the causal left halo
  if (tid < LC_HALO) tile[tid] = 0.0f;

#if LC_HAVE_ASYNC
  {
    typedef __attribute__((address_space(1))) v4i gv4i_t;
    typedef __attribute__((address_space(3))) v4i lv4i_t;
    gv4i_t* gsrc = (gv4i_t*)xrow;                 // builtin reads only; non-const sig
    lv4i_t* ldst = (lv4i_t*)(tile + LC_HALO);
#pragma unroll
    for (int j = 0; j < LC_ITERS; ++j) {
      const int i = tid + j * LC_THREADS;         // 16B-unit index, lane-contiguous
      __builtin_amdgcn_global_load_async_to_lds_b128(gsrc + i, ldst + i, 0, 0);
    }
  }
  lc_wait_async();
#else
  {
    const v4f* xv = (const v4f*)xrow;
#pragma unroll
    for (int j = 0; j < LC_ITERS; ++j) {
      const int i = tid + j * LC_THREADS;
      v4f v = xv[i];
      *(v4f*)(tile + LC_HALO + i * 4) = v;
    }
  }
#endif
  __syncthreads();

  // per-head softmaxed taps (uniform per block -> scalar loads)
  float wk[LC_K];
#pragma unroll
  for (int k = 0; k < LC_K; ++k) wk[k] = sw[h * LC_K + k];
  const float bb = bias[h];

#pragma unroll
  for (int j = 0; j < LC_ITERS; ++j) {
    const int t0 = (tid + j * LC_THREADS) * 4;  // 4 contiguous outputs
    // window covers x[t0-8 .. t0+3]; 16B-aligned, lane stride 16B (conflict-free)
    const v4f* wptr = (const v4f*)(tile + LC_HALO + t0 - 8);
    v4f va = wptr[0];
    v4f vb = wptr[1];
    v4f vc = wptr[2];
    float win[12] = {va.x, va.y, va.z, va.w, vb.x, vb.y, vb.z, vb.w,
                     vc.x, vc.y, vc.z, vc.w};
    // out[t0+i] = bb + sum_k wk[k] * x[t0+i+k-6]  -> win[i+2+k]
    float a0 = bb, a1 = bb, a2 = bb, a3 = bb;
#pragma unroll
    for (int k = 0; k < LC_K; ++k) {
      a0 = fmaf(wk[k], win[2 + k], a0);
      a1 = fmaf(wk[k], win[3 + k], a1);
      a2 = fmaf(wk[k], win[4 + k], a2);
      a3 = fmaf(wk[k], win[5 + k], a3);
    }
    v4f r;
    r.x = a0; r.y = a1; r.z = a2; r.w = a3;
    __builtin_nontemporal_store(r, (v4f*)(orow + t0));  // streaming output
  }
}

extern "C" void kernel_launch(void* const* d_in, const int* in_sizes, int n_in,
                              void* d_out, int out_size, void* d_ws, size_t ws_size,
                              hipStream_t stream) {
  (void)in_sizes; (void)n_in; (void)out_size; (void)ws_size;
  const float* x    = (const float*)d_in[0];  // [B, C, T]
  const float* w    = (const float*)d_in[1];  // [H, 1, K]
  const float* bias = (const float*)d_in[2];  // [H]
  float* out        = (float*)d_out;          // [B, C, T]
  float* sw         = (float*)d_ws;           // H*K floats of scratch

  lc_softmax_kernel<<<1, 32, 0, stream>>>(w, sw);
  lc_conv_kernel<<<LC_B * LC_C, LC_THREADS, 0, stream>>>(x, sw, bias, out);
}